// FM_75557064671821
// MI455X (gfx1250) — compile-verified
//
#include <hip/hip_runtime.h>
#include <math.h>

typedef __attribute__((ext_vector_type(2))) float v2f;
typedef __attribute__((ext_vector_type(8))) float v8f;

#define EMB 16
#define NFIELD 4
#define SPW 16                 // samples per wave (WMMA tile M)
#define WPB 4                  // waves per block
#define THREADS (WPB * 32)
#define LSTRIDE 17             // padded row stride -> conflict-free LDS banks

// field offsets: cumsum of [31360, 6807, 18, 94]
#define OFF1 31360
#define OFF2 38167
#define OFF3 38185

__global__ __launch_bounds__(THREADS)
void fm_fwd_kernel(const int* __restrict__ x, const float* __restrict__ v,
                   const float* __restrict__ bias, float* __restrict__ out)
{
    __shared__ float S_lds[WPB][SPW * LSTRIDE];   // summed embeddings, [sample][k]
    __shared__ float R_lds[WPB][SPW];             // sum_k e^2 - 2*sum_f bias

    const int lane = threadIdx.x & 31;
    const int wave = threadIdx.x >> 5;
    const int waveBase = (blockIdx.x * WPB + wave) * SPW;

    float* S = S_lds[wave];
    float* R = R_lds[wave];

    const int h = lane >> 4;   // half-wave id
    const int k = lane & 15;   // embedding dim handled by this lane

    // ---- gather phase: 2 samples per iteration (one per half-wave) ----
    #pragma unroll
    for (int p = 0; p < SPW / 2; ++p) {
        const int m = 2 * p + h;             // sample index within wave
        const int s = waveBase + m;
        const int4 xi = ((const int4*)x)[s]; // 16B, same addr across half-wave -> L2 broadcast
        int xo[NFIELD];
        xo[0] = xi.x;
        xo[1] = xi.y + OFF1;
        xo[2] = xi.z + OFF2;
        xo[3] = xi.w + OFF3;

        float ssum = 0.0f, r = 0.0f;
        #pragma unroll
        for (int f = 0; f < NFIELD; ++f) {
            float e = v[xo[f] * EMB + k];    // 64B coalesced per half-wave
            ssum += e;
            r += e * e;
        }
        // fold bias into the reduced term: result = 0.5*(diag - r_reduced)
        if (k < NFIELD) r -= 2.0f * bias[xo[k]];

        // reduce r across the 16 lanes of this half-wave
        #pragma unroll
        for (int off = 8; off > 0; off >>= 1)
            r += __shfl_xor(r, off, 16);

        S[m * LSTRIDE + k] = ssum;
        if (k == 0) R[m] = r;
    }

    __syncthreads();

    // ---- WMMA phase: D = S * S^T  (16x16 f32), K accumulated in chunks of 4 ----
    // A (16x4 f32) layout: lanes 0-15: M=lane, VGPR0->K=0, VGPR1->K=1;
    //                      lanes 16-31: M=lane-16, VGPR0->K=2, VGPR1->K=3.
    // B (4x16 f32) layout: VGPR0: lanes 0-15 K=0, lanes 16-31 K=1;
    //                      VGPR1: lanes 0-15 K=2, lanes 16-31 K=3.  B[k][n] = S[n][k].
    const int am = lane & 15;
    const int ak = (lane >> 4) << 1;   // 0 or 2
    const int bn = lane & 15;
    const int bk = lane >> 4;          // 0 or 1

    v8f acc = {};
    #pragma unroll
    for (int c = 0; c < 4; ++c) {
        const int kc = 4 * c;
        v2f a, b;
        a.x = S[am * LSTRIDE + kc + ak];
        a.y = S[am * LSTRIDE + kc + ak + 1];
        b.x = S[bn * LSTRIDE + kc + bk];
        b.y = S[bn * LSTRIDE + kc + bk + 2];
        acc = __builtin_amdgcn_wmma_f32_16x16x4_f32(
            /*neg_a=*/false, a, /*neg_b=*/false, b,
            /*c_mod=*/(short)0, acc, /*reuse_a=*/false, /*reuse_b=*/false);
    }

    // ---- extract diag(D): D[m][m] at (VGPR m, lane m) for m<8,
    //      (VGPR m-8, lane m+16) for m>=8 ----
    float diag = 0.0f;
    int m = -1;
    if (lane < 8)       m = lane;
    else if (lane >= 24) m = lane - 16;
    #pragma unroll
    for (int i = 0; i < 8; ++i) {
        float di = acc[i];
        if (lane == i)      diag = di;
        if (lane == 24 + i) diag = di;
    }

    if (m >= 0) {
        float z = 0.5f * (diag - R[m]);
        out[waveBase + m] = 1.0f / (1.0f + __expf(-z));
    }
}

extern "C" void kernel_launch(void* const* d_in, const int* in_sizes, int n_in,
                              void* d_out, int out_size, void* d_ws, size_t ws_size,
                              hipStream_t stream) {
    const int*   x    = (const int*)d_in[0];    // (4096, 4) int32
    const float* v    = (const float*)d_in[1];  // (38279, 16) f32
    const float* bias = (const float*)d_in[2];  // (38279, 1) f32
    float*       out  = (float*)d_out;          // (4096, 1) f32

    const int batch = in_sizes[0] / NFIELD;     // 4096
    const int samplesPerBlock = WPB * SPW;      // 64
    const int grid = batch / samplesPerBlock;   // 64 blocks (batch is a multiple of 64)

    fm_fwd_kernel<<<grid, THREADS, 0, stream>>>(x, v, bias, out);
}